// ComputeDepthMaps_23424751633217
// MI455X (gfx1250) — compile-verified
//
#include <hip/hip_runtime.h>
#include <math.h>
#include <limits.h>

typedef float v2f __attribute__((ext_vector_type(2)));
typedef float v8f __attribute__((ext_vector_type(8)));

#define IMG 256
#define IMG2 (IMG * IMG)

// ds_swizzle group-of-32 XOR exchange: offset = xor_mask<<10 | and_mask(0x1F)
#define SWZ_XOR(v, m) \
    __int_as_float(__builtin_amdgcn_ds_swizzle(__float_as_int(v), (((m) << 10) | 0x1F)))

// Workspace layout (as float/int words):
//   ws[0..15]  : M (4x4 row-major), written by init kernel
//   wsi[16]    : zmin as monotone int key (init INT_MAX)
//   wsi[17]    : zmax as monotone int key (init INT_MIN)
//   ws[32 .. 32+N)       : pts_i per point
//   ws[32+N .. 32+2N)    : pts_j per point
//   ws[32+2N .. 32+3N)   : z (ndc depth) per point

__device__ __forceinline__ int float_key(float f) {
    int k = __float_as_int(f);
    return (k >= 0) ? k : (k ^ 0x7FFFFFFF);   // monotone: int order == float order
}
__device__ __forceinline__ float key_float(int k) {
    return __int_as_float((k >= 0) ? k : (k ^ 0x7FFFFFFF));
}

// ---------------------------------------------------------------------------
// Kernel 1: zero the canvas; thread 0 builds the proj*view matrix from view_id
// and initializes the z min/max reduction slots.
// ---------------------------------------------------------------------------
__global__ void cdm_init_kernel(const int* __restrict__ viewPtr,
                                float* __restrict__ out, int outN,
                                float* __restrict__ ws) {
    int idx = blockIdx.x * blockDim.x + threadIdx.x;
    for (int i = idx; i < outN; i += gridDim.x * blockDim.x) out[i] = 0.0f;

    if (idx == 0) {
        int vid = viewPtr[0];
        if (vid < 0) vid = 0;
        if (vid > 7) vid = 7;
        // EYES_POS_LIST[vid] = (+-1,+-1,+-1) in binary-count order
        float ex = (vid & 4) ? 1.0f : -1.0f;
        float ey = (vid & 2) ? 1.0f : -1.0f;
        float ez = (vid & 1) ? 1.0f : -1.0f;
        // z axis = normalize(eyes)
        float n = fmaxf(sqrtf(ex * ex + ey * ey + ez * ez), 1e-6f);
        float zx = ex / n, zy = ey / n, zz = ez / n;
        // x = normalize(cross(up=(0,0,1), z)) = normalize(-zy, zx, 0)
        float xx = -zy, xy = zx, xz = 0.0f;
        float xn = fmaxf(sqrtf(xx * xx + xy * xy), 1e-6f);
        xx /= xn; xy /= xn;
        // y = cross(z, x)
        float yx = zy * xz - zz * xy;
        float yy = zz * xx - zx * xz;
        float yz = zx * xy - zy * xx;
        // view = orient @ trans  (rows x,y,z ; translation = -R*eyes)
        float V[4][4] = {
            { xx, xy, xz, -(xx * ex + xy * ey + xz * ez) },
            { yx, yy, yz, -(yx * ex + yy * ey + yz * ez) },
            { zx, zy, zz, -(zx * ex + zy * ey + zz * ez) },
            { 0.f, 0.f, 0.f, 1.f } };
        const float zn = 0.1f, zf = 10.0f;
        const float p22 = -2.0f / (zf - zn);
        const float p23 = (zf + zn) / (zf - zn);
        for (int c = 0; c < 4; ++c) {
            ws[0 * 4 + c] = 1.5f * V[0][c];
            ws[1 * 4 + c] = 1.5f * V[1][c];
            ws[2 * 4 + c] = p22 * V[2][c] + p23 * V[3][c];
            ws[3 * 4 + c] = V[3][c];
        }
        int* wsi = (int*)ws;
        wsi[16] = INT_MAX;   // zmin key
        wsi[17] = INT_MIN;   // zmax key
    }
}

// ---------------------------------------------------------------------------
// Kernel 2: WMMA point transform.  D(16x16) = M_ext(16x4) x P(4x16).
//   A (16x4 f32, 2 VGPRs): lane L<16  -> {M[L][0], M[L][1]}   (rows>=4 zero)
//                          lane L>=16 -> {M[L-16][2], M[L-16][3]}
//   B (4x16 f32, 2 VGPRs): lane L<16  -> {x_L, y_L}
//                          lane L>=16 -> {z_(L-16), 1.0}
//   D: lane L<16, VGPR m -> hom component m of point L  (rows 8..15 unused)
// Then perspective divide (fast rcp: hom_w == 1 by construction), pixel
// coords, and a ds_swizzle-based wave min/max reduction of z.
// ---------------------------------------------------------------------------
__global__ void cdm_transform_kernel(const float* __restrict__ data,
                                     float* __restrict__ ws, int N) {
    const int lane = threadIdx.x & 31;
    const int wid  = threadIdx.x >> 5;
    const int wavesPerBlock = blockDim.x >> 5;
    const int ptBase = (blockIdx.x * wavesPerBlock + wid) * 16;
    if (ptBase >= N) return;              // wave-uniform: EXEC stays all-ones

    const int lp = lane & 15;
    const int pt = ptBase + lp;

    // B operand: homogeneous point coords
    const float* P = data + 3 * pt;
    v2f b;
    if (lane < 16) { b.x = P[0]; b.y = P[1]; }
    else           { b.x = P[2]; b.y = 1.0f; }

    // A operand: M extended to 16 rows (rows >= 4 are zero)
    v2f a; a.x = 0.0f; a.y = 0.0f;
    if (lp < 4) {
        if (lane < 16) { a.x = ws[lp * 4 + 0]; a.y = ws[lp * 4 + 1]; }
        else           { a.x = ws[lp * 4 + 2]; a.y = ws[lp * 4 + 3]; }
    }

    v8f c = {};
    c = __builtin_amdgcn_wmma_f32_16x16x4_f32(
            /*neg_a=*/false, a, /*neg_b=*/false, b,
            /*c_mod=*/(short)0, c, /*reuse_a=*/false, /*reuse_b=*/false);

    // lanes 0..15: c[0..3] = hom x,y,z,w of point pt.  hom_w == 1 exactly
    // (M row 3 = [0,0,0,1]) so raw v_rcp_f32 is exact here.
    float inv = __builtin_amdgcn_rcpf(c[3]);
    float xn = c[0] * inv;
    float yn = c[1] * inv;
    float zn = c[2] * inv;
    float pi = (-yn * 0.5f + 0.5f) * (float)(IMG - 1);
    float pj = ( xn * 0.5f + 0.5f) * (float)(IMG - 1);

    // wave-wide min/max of z via ds_swizzle xor exchanges
    // (lanes >= 16 hold garbage -> neutral seeds)
    float zmin = (lane < 16) ? zn :  INFINITY;
    float zmax = (lane < 16) ? zn : -INFINITY;
    zmin = fminf(zmin, SWZ_XOR(zmin, 16));
    zmax = fmaxf(zmax, SWZ_XOR(zmax, 16));
    zmin = fminf(zmin, SWZ_XOR(zmin, 8));
    zmax = fmaxf(zmax, SWZ_XOR(zmax, 8));
    zmin = fminf(zmin, SWZ_XOR(zmin, 4));
    zmax = fmaxf(zmax, SWZ_XOR(zmax, 4));
    zmin = fminf(zmin, SWZ_XOR(zmin, 2));
    zmax = fmaxf(zmax, SWZ_XOR(zmax, 2));
    zmin = fminf(zmin, SWZ_XOR(zmin, 1));
    zmax = fmaxf(zmax, SWZ_XOR(zmax, 1));
    if (lane == 0) {
        int* wsi = (int*)ws;
        atomicMin(&wsi[16], float_key(zmin));
        atomicMax(&wsi[17], float_key(zmax));
    }

    if (lane < 16) {
        float* pts_i = ws + 32;
        float* pts_j = pts_i + N;
        float* zarr  = pts_j + N;
        pts_i[pt] = pi;
        pts_j[pt] = pj;
        zarr[pt]  = zn;
    }
}

// ---------------------------------------------------------------------------
// Kernel 3: splat.  One wave per point; lane owns one dj column of the 22x22
// window, loops over 22 rows.  Values are >= 0, so unsigned-bit-pattern
// atomicMax == float max against a zero-initialized canvas (= max(0, canvas)).
// ---------------------------------------------------------------------------
__global__ void cdm_splat_kernel(const float* __restrict__ ws,
                                 float* __restrict__ out,
                                 int N, int nPerBatch, int batchShift) {
    const int lane = threadIdx.x & 31;
    const int wave = blockIdx.x * (blockDim.x >> 5) + (threadIdx.x >> 5);
    if (wave >= N) return;                 // wave-uniform
    const int pt = wave;

    const float* pts_i = ws + 32;
    const float* pts_j = pts_i + N;
    const float* zarr  = pts_j + N;
    const int*   wsi   = (const int*)ws;

    const float zmin = key_float(wsi[16]);
    const float zmax = key_float(wsi[17]);

    const float pi = pts_i[pt];
    const float pj = pts_j[pt];
    const float z  = zarr[pt];
    const float feat = 1.0f - (z - zmin) / (zmax - zmin);

    const int batch = (batchShift >= 0) ? (pt >> batchShift) : (pt / nPerBatch);
    const int bi = (int)floorf(pi);
    const int bj = (int)floorf(pj);

    // lane -> dj in [-10, 11]; lanes 22..31 idle
    const int dj  = lane - 10;
    const int pjj = bj + dj;
    const bool jok = (lane < 22) & (pjj >= 0) & (pjj < IMG);
    const float dy  = (float)pjj - pj;
    const float dy2 = dy * dy;

    unsigned int* outu = (unsigned int*)out;
    const int rowBase = batch * IMG2 + pjj;

    #pragma unroll
    for (int di = -10; di <= 11; ++di) {
        const int pii = bi + di;
        if (pii < 0 || pii >= IMG) continue;         // wave-uniform branch
        const float dx = (float)pii - pi;
        const float d2 = dx * dx + dy2;
        if (jok && d2 <= 100.0f) {
            const float d = sqrtf(d2);
            const float w = __cosf(d * 0.15707963267948966f);  // pi/(2*radius)
            const float val = w * feat;                         // >= 0
            atomicMax(&outu[rowBase + pii * IMG], __float_as_uint(val));
        }
    }
}

extern "C" void kernel_launch(void* const* d_in, const int* in_sizes, int n_in,
                              void* d_out, int out_size, void* d_ws, size_t ws_size,
                              hipStream_t stream) {
    const float* data    = (const float*)d_in[0];
    const int*   viewPtr = (const int*)d_in[1];
    float* out = (float*)d_out;
    float* ws  = (float*)d_ws;

    const int N = in_sizes[0] / 3;          // total points (32*4096)
    const int B = out_size / IMG2;          // batches
    const int nPerBatch = N / B;
    // power-of-two fast path for the batch index
    int batchShift = -1;
    if (nPerBatch > 0 && (nPerBatch & (nPerBatch - 1)) == 0) {
        batchShift = 0;
        while ((1 << batchShift) != nPerBatch) ++batchShift;
    }

    // 1) zero canvas + build matrix + init reductions
    cdm_init_kernel<<<1024, 256, 0, stream>>>(viewPtr, out, out_size, ws);

    // 2) WMMA transform: 8 waves * 16 points = 128 points / block
    const int tBlocks = (N + 127) / 128;
    cdm_transform_kernel<<<tBlocks, 256, 0, stream>>>(data, ws, N);

    // 3) splat: one wave per point, 8 waves / block
    const int sBlocks = (N + 7) / 8;
    cdm_splat_kernel<<<sBlocks, 256, 0, stream>>>(ws, out, N, nPerBatch, batchShift);
}